// ClusteringLayer_26688926778097
// MI455X (gfx1250) — compile-verified
//
#include <hip/hip_runtime.h>

// ---------------------------------------------------------------------------
// Greedy per-cacheline clustering (THRESHOLD = 0.1, 64 f32 per line).
// One thread owns one line in registers; lines are staged block-wide through
// LDS with the CDNA5 async global<->LDS DMA path.
// ---------------------------------------------------------------------------

#define THREADS          256
#define LINES_PER_BLOCK  256
#define LINE             64
#define CHUNKS_PER_LINE  16          // 16 x float4 = 256 B per line
#define LDS_STRIDE_F     68          // 272 B per line: 16B-aligned, bank-padded

// Builtin expects int4 pointers with explicit address spaces
// (clang prints addrspace(1) as "__device__" in HIP diagnostics).
typedef int v4i __attribute__((ext_vector_type(4)));
typedef __attribute__((address_space(1))) v4i* g4_t;   // global int4*
typedef __attribute__((address_space(3))) v4i* l4_t;   // LDS    int4*

__device__ __forceinline__ void wait_asynccnt0() {
#if __has_builtin(__builtin_amdgcn_s_wait_asynccnt)
  __builtin_amdgcn_s_wait_asynccnt(0);
#else
  asm volatile("s_wait_asynccnt 0" ::: "memory");
#endif
}

// Exact reimplementation of the reference recurrence:
//  - comparing against post-snap values without the is_base mask is identical
//    (a non-base is an exact copy of an earlier base, which matches earlier);
//  - descending-j unconditional select == "first (smallest-j) match wins".
// Inner body lowers to v_sub_f32 + v_cmp_lt_f32(|.|) + v_cndmask_b32.
__device__ __forceinline__ void cluster_line_regs(float v[LINE]) {
#pragma unroll
  for (int i = 1; i < LINE; ++i) {
    const float xi = v[i];
    float val = xi;
#pragma unroll
    for (int j = i - 1; j >= 0; --j) {
      const float d = v[j] - xi;
      val = (__builtin_fabsf(d) < 0.1f) ? v[j] : val;
    }
    v[i] = val;
  }
}

__global__ __launch_bounds__(THREADS) void cluster_lines_async(
    const float* __restrict__ src, float* __restrict__ dst) {
  __shared__ float lds[LINES_PER_BLOCK * LDS_STRIDE_F];  // 69,632 B
  const int t = threadIdx.x;
  const long long base = (long long)blockIdx.x * (LINES_PER_BLOCK * LINE);

  // ---- stage: global -> LDS, coalesced 16B chunks via async DMA -----------
#pragma unroll
  for (int k = 0; k < (LINES_PER_BLOCK * CHUNKS_PER_LINE) / THREADS; ++k) {
    const int c    = k * THREADS + t;   // chunk id within block (coalesced)
    const int line = c >> 4;
    const int q    = c & 15;
    const float* g = src + base + (long long)c * 4;
    float*       l = &lds[line * LDS_STRIDE_F + q * 4];
    __builtin_amdgcn_global_load_async_to_lds_b128((g4_t)g, (l4_t)l, 0, 0);
  }
  wait_asynccnt0();     // this wave's async loads landed in LDS
  __syncthreads();      // all waves' loads visible

  // ---- pull own line into registers (ds_load_b128, 16B aligned) -----------
  float v[LINE];
  float* myl = &lds[t * LDS_STRIDE_F];
#pragma unroll
  for (int k = 0; k < CHUNKS_PER_LINE; ++k) {
    const float4 q4 = *(const float4*)(myl + k * 4);
    v[k * 4 + 0] = q4.x; v[k * 4 + 1] = q4.y;
    v[k * 4 + 2] = q4.z; v[k * 4 + 3] = q4.w;
  }

  cluster_line_regs(v);

  // Own line only -> no barrier needed before overwriting it in LDS.
#pragma unroll
  for (int k = 0; k < CHUNKS_PER_LINE; ++k) {
    float4 q4;
    q4.x = v[k * 4 + 0]; q4.y = v[k * 4 + 1];
    q4.z = v[k * 4 + 2]; q4.w = v[k * 4 + 3];
    *(float4*)(myl + k * 4) = q4;
  }
  __syncthreads();      // all results in LDS before cross-line readout

  // ---- drain: LDS -> global, coalesced async stores -----------------------
#pragma unroll
  for (int k = 0; k < (LINES_PER_BLOCK * CHUNKS_PER_LINE) / THREADS; ++k) {
    const int c    = k * THREADS + t;
    const int line = c >> 4;
    const int q    = c & 15;
    float* g = dst + base + (long long)c * 4;
    float* l = &lds[line * LDS_STRIDE_F + q * 4];
    __builtin_amdgcn_global_store_async_from_lds_b128((g4_t)g, (l4_t)l, 0, 0);
  }
  wait_asynccnt0();     // ordered before endpgm (endpgm also waits idle)
}

// Fallback for a partial block of lines (not hit for this shape, kept exact).
__global__ void cluster_tail(const float* __restrict__ src,
                             float* __restrict__ dst,
                             long long line0, int nlines) {
  const int i = blockIdx.x * blockDim.x + threadIdx.x;
  if (i >= nlines) return;
  const long long b = (line0 + i) * (long long)LINE;
  float v[LINE];
#pragma unroll
  for (int k = 0; k < CHUNKS_PER_LINE; ++k) {
    const float4 q4 = *(const float4*)(src + b + k * 4);
    v[k * 4 + 0] = q4.x; v[k * 4 + 1] = q4.y;
    v[k * 4 + 2] = q4.z; v[k * 4 + 3] = q4.w;
  }
  cluster_line_regs(v);
#pragma unroll
  for (int k = 0; k < CHUNKS_PER_LINE; ++k) {
    float4 q4;
    q4.x = v[k * 4 + 0]; q4.y = v[k * 4 + 1];
    q4.z = v[k * 4 + 2]; q4.w = v[k * 4 + 3];
    *(float4*)(dst + b + k * 4) = q4;
  }
}

// Elements past the last full line are passed through unchanged (n % 64 == 0
// for this shape, so this normally never launches).
__global__ void copy_tail(const float* __restrict__ src, float* __restrict__ dst,
                          long long off, long long n) {
  const long long i = (long long)blockIdx.x * blockDim.x + threadIdx.x;
  if (i < n) dst[off + i] = src[off + i];
}

extern "C" void kernel_launch(void* const* d_in, const int* in_sizes, int n_in,
                              void* d_out, int out_size, void* d_ws, size_t ws_size,
                              hipStream_t stream) {
  const float* x = (const float*)d_in[0];
  float* out = (float*)d_out;

  const long long n        = (long long)in_sizes[0];    // 51,380,224
  const long long n_lines  = n / LINE;                  // 802,816
  const long long n_full   = n_lines / LINES_PER_BLOCK; // 3,136 blocks
  const long long rem_ln   = n_lines - n_full * LINES_PER_BLOCK;
  const long long tail_el  = n - n_lines * LINE;

  if (n_full > 0) {
    cluster_lines_async<<<(int)n_full, THREADS, 0, stream>>>(x, out);
  }
  if (rem_ln > 0) {
    cluster_tail<<<(int)((rem_ln + 255) / 256), 256, 0, stream>>>(
        x, out, n_full * LINES_PER_BLOCK, (int)rem_ln);
  }
  if (tail_el > 0) {
    copy_tail<<<(int)((tail_el + 255) / 256), 256, 0, stream>>>(
        x, out, n_lines * LINE, tail_el);
  }
}